// R_Mamba_62852551410077
// MI455X (gfx1250) — compile-verified
//
#include <hip/hip_runtime.h>
#include <hip/hip_bf16.h>
#include <math.h>

// Problem constants (match reference)
#define B_    2
#define INC_  64
#define HID_  96
#define H_    96
#define W_    96
#define L_    (H_*W_)        // 9216
#define DI_   128
#define NS_   16             // state dim N
#define RR_   4              // dt rank R
#define KD_   4              // scan directions K
#define CDIM_ 36             // R + 2N
#define EPS_  1e-5f
#define CS_   128            // scan chunk size
#define NC_   (L_/CS_)       // 72 chunks

typedef __attribute__((ext_vector_type(2))) float v2f;
typedef __attribute__((ext_vector_type(8))) float v8f;

// ---- CDNA5 WMMA f32 16x16x4 (D = A(16x4) * B(4x16) + C) ----
// A: lane(0..15)=row M, VGPR j -> K = 2*(lane>=16)+j  (ISA 7.12.2, 32-bit A 16x4)
// B: lane(0..15)=col N, VGPR j -> K = 2*(lane>=16)+j
// C/D: VGPR i -> M = 8*(lane>=16)+i, N = lane&15
__device__ __forceinline__ v8f wmma_f32_k4(v2f a, v2f b, v8f c) {
  return __builtin_amdgcn_wmma_f32_16x16x4_f32(false, a, false, b, (short)0, c,
                                               false, false);
}

__device__ __forceinline__ float sigmoidf_(float x) { return 1.0f / (1.0f + __expf(-x)); }
__device__ __forceinline__ float siluf_(float x)    { return x * sigmoidf_(x); }
__device__ __forceinline__ float softplusf_(float x){ return (x > 20.0f) ? x : __logf(1.0f + __expf(x)); }

// scan position t of direction k -> original hw pixel index
__device__ __forceinline__ int pixmap(int k, int t) {
  int l = (k >= 2) ? (L_ - 1 - t) : t;
  if (k & 1) { int wi = l / H_; int hi = l - wi * H_; return hi * W_ + wi; }
  return l;
}

// Cooperative async stage of `rows` x CS_ floats of xdbl (row stride L_) into LDS.
// Uses gfx1250 GLOBAL_LOAD_ASYNC_TO_LDS_B128 (ASYNCcnt), then drains the counter.
// Caller must __syncthreads() before reading.
__device__ __forceinline__ void stage_tile_async(float* sdst, const float* gsrc, int rows) {
  // rows*CS_/4 b128 transfers; 128 threads cooperate
  const int total = rows * (CS_ / 4);
  for (int j = threadIdx.x; j < total; j += 128) {
    const int r  = j >> 5;           // CS_/4 == 32 x b128 per row
    const int c4 = j & 31;
    unsigned int lds_off = (unsigned int)(size_t)(sdst + r * CS_ + c4 * 4);
    const float* g = gsrc + (size_t)r * L_ + c4 * 4;
    asm volatile("global_load_async_to_lds_b128 %0, %1, off"
                 :: "v"(lds_off), "v"(g) : "memory");
  }
  asm volatile("s_wait_asynccnt 0" ::: "memory");
}

// ======================= conv_bn_silu: out[b,o,l] = silu((W x + b)*g/sqrt(1+eps)+beta)
// Each wave: one 16-row o-tile x four 16-col l-tiles; A fragment reused 4x.
__global__ void k_conv_bn_silu(const float* __restrict__ x, const float* __restrict__ w,
                               const float* __restrict__ bias, const float* __restrict__ g,
                               const float* __restrict__ beta, float* __restrict__ out) {
  const int lane = threadIdx.x & 31, wave = threadIdx.x >> 5;
  const int half = lane >> 4, l16 = lane & 15;
  const int n0 = (blockIdx.x * 4 + wave) * 64;   // 4 l tiles
  const int m0 = blockIdx.y * 16;                // o tile
  const int b  = blockIdx.z;
  const float* xb = x + (size_t)b * INC_ * L_;
  v8f acc0 = {}, acc1 = {}, acc2 = {}, acc3 = {};
  for (int k0 = 0; k0 < INC_; k0 += 4) {
    const int ka = k0 + 2 * half;
    v2f a;
    a.x = w[(m0 + l16) * INC_ + ka];
    a.y = w[(m0 + l16) * INC_ + ka + 1];
    const float* r0 = xb + (size_t)ka * L_ + n0 + l16;
    const float* r1 = xb + (size_t)(ka + 1) * L_ + n0 + l16;
    v2f b0, b1, b2, b3;
    b0.x = r0[0];  b0.y = r1[0];
    b1.x = r0[16]; b1.y = r1[16];
    b2.x = r0[32]; b2.y = r1[32];
    b3.x = r0[48]; b3.y = r1[48];
    acc0 = wmma_f32_k4(a, b0, acc0);
    acc1 = wmma_f32_k4(a, b1, acc1);
    acc2 = wmma_f32_k4(a, b2, acc2);
    acc3 = wmma_f32_k4(a, b3, acc3);
  }
  const float inv = rsqrtf(1.0f + EPS_);
  v8f* accs[4] = { &acc0, &acc1, &acc2, &acc3 };
#pragma unroll
  for (int jt = 0; jt < 4; ++jt) {
    v8f& a = *accs[jt];
#pragma unroll
    for (int i = 0; i < 8; ++i) {
      int m = m0 + 8 * half + i;
      float v = (a[i] + bias[m]) * inv * g[m] + beta[m];
      out[((size_t)b * HID_ + m) * L_ + n0 + 16 * jt + l16] = siluf_(v);
    }
  }
}

// ======================= patch_embed GEMM: yraw[b,l,o] = sum_c t[b,c,l] w[o,c] + pb[o]
// Four l(M)-tiles per wave; B (weight) fragment reused 4x.
__global__ void k_pe_gemm(const float* __restrict__ t, const float* __restrict__ w,
                          const float* __restrict__ pb, float* __restrict__ yraw) {
  const int lane = threadIdx.x & 31, wave = threadIdx.x >> 5;
  const int half = lane >> 4, l16 = lane & 15;
  const int m0 = (blockIdx.x * 4 + wave) * 64;   // 4 l tiles
  const int n0 = blockIdx.y * 16;                // o tile
  const int b  = blockIdx.z;
  const float* tb = t + (size_t)b * HID_ * L_;
  v8f acc0 = {}, acc1 = {}, acc2 = {}, acc3 = {};
  for (int k0 = 0; k0 < HID_; k0 += 4) {
    const int ka = k0 + 2 * half;
    v2f bb;
    bb.x = w[(n0 + l16) * HID_ + ka];
    bb.y = w[(n0 + l16) * HID_ + ka + 1];
    const float* r0 = tb + (size_t)ka * L_ + m0 + l16;
    const float* r1 = tb + (size_t)(ka + 1) * L_ + m0 + l16;
    v2f a0, a1, a2, a3;
    a0.x = r0[0];  a0.y = r1[0];
    a1.x = r0[16]; a1.y = r1[16];
    a2.x = r0[32]; a2.y = r1[32];
    a3.x = r0[48]; a3.y = r1[48];
    acc0 = wmma_f32_k4(a0, bb, acc0);
    acc1 = wmma_f32_k4(a1, bb, acc1);
    acc2 = wmma_f32_k4(a2, bb, acc2);
    acc3 = wmma_f32_k4(a3, bb, acc3);
  }
  v8f* accs[4] = { &acc0, &acc1, &acc2, &acc3 };
#pragma unroll
  for (int jt = 0; jt < 4; ++jt) {
    v8f& a = *accs[jt];
#pragma unroll
    for (int i = 0; i < 8; ++i) {
      int l = m0 + 16 * jt + 8 * half + i;
      yraw[((size_t)b * L_ + l) * HID_ + n0 + l16] = a[i] + pb[n0 + l16];
    }
  }
}

// ======================= LayerNorm over 96 (one wave per row, in place)
__global__ void k_ln96(float* __restrict__ y, const float* __restrict__ g,
                       const float* __restrict__ bta, int rows) {
  int row = blockIdx.x * (blockDim.x >> 5) + (threadIdx.x >> 5);
  if (row >= rows) return;
  int lane = threadIdx.x & 31;
  float* p = y + (size_t)row * HID_;
  float v0 = p[lane], v1 = p[lane + 32], v2 = p[lane + 64];
  float s = v0 + v1 + v2;
  for (int o = 16; o >= 1; o >>= 1) s += __shfl_xor(s, o, 32);
  float mu = s * (1.0f / 96.0f);
  float d0 = v0 - mu, d1 = v1 - mu, d2 = v2 - mu;
  float q = d0 * d0 + d1 * d1 + d2 * d2;
  for (int o = 16; o >= 1; o >>= 1) q += __shfl_xor(q, o, 32);
  float inv = rsqrtf(q * (1.0f / 96.0f) + EPS_);
  p[lane]      = d0 * inv * g[lane]      + bta[lane];
  p[lane + 32] = d1 * inv * g[lane + 32] + bta[lane + 32];
  p[lane + 64] = d2 * inv * g[lane + 64] + bta[lane + 64];
}

// ======================= in_proj GEMM: xz[b,e,l] = sum_c y[b,l,c] w[e,c]
__global__ void k_inproj(const float* __restrict__ y, const float* __restrict__ w,
                         float* __restrict__ xz) {
  const int lane = threadIdx.x & 31, wave = threadIdx.x >> 5;
  const int half = lane >> 4, l16 = lane & 15;
  const int n0 = (blockIdx.x * 4 + wave) * 64;   // 4 l tiles
  const int m0 = blockIdx.y * 16;                // e tile (256/16)
  const int b  = blockIdx.z;
  v8f acc0 = {}, acc1 = {}, acc2 = {}, acc3 = {};
  for (int k0 = 0; k0 < HID_; k0 += 4) {
    const int ka = k0 + 2 * half;
    v2f a;
    a.x = w[(m0 + l16) * HID_ + ka];
    a.y = w[(m0 + l16) * HID_ + ka + 1];
    const float* yb = y + ((size_t)b * L_ + n0 + l16) * HID_ + ka;
    v2f b0, b1, b2, b3;
    b0.x = yb[0];         b0.y = yb[1];
    b1.x = yb[16 * HID_]; b1.y = yb[16 * HID_ + 1];
    b2.x = yb[32 * HID_]; b2.y = yb[32 * HID_ + 1];
    b3.x = yb[48 * HID_]; b3.y = yb[48 * HID_ + 1];
    acc0 = wmma_f32_k4(a, b0, acc0);
    acc1 = wmma_f32_k4(a, b1, acc1);
    acc2 = wmma_f32_k4(a, b2, acc2);
    acc3 = wmma_f32_k4(a, b3, acc3);
  }
  v8f* accs[4] = { &acc0, &acc1, &acc2, &acc3 };
#pragma unroll
  for (int jt = 0; jt < 4; ++jt) {
    v8f& a = *accs[jt];
#pragma unroll
    for (int i = 0; i < 8; ++i) {
      int e = m0 + 8 * half + i;
      xz[((size_t)b * 2 * DI_ + e) * L_ + n0 + 16 * jt + l16] = a[i];
    }
  }
}

// ======================= depthwise 3x3 + bias + silu (SAME, zero pad)
__global__ void k_dwconv(const float* __restrict__ xz, const float* __restrict__ dww,
                         const float* __restrict__ dwb, float* __restrict__ xc) {
  size_t idx = (size_t)blockIdx.x * blockDim.x + threadIdx.x;
  if (idx >= (size_t)B_ * DI_ * L_) return;
  int l = (int)(idx % L_);
  size_t tmp = idx / L_;
  int d = (int)(tmp % DI_);
  int b = (int)(tmp / DI_);
  int h = l / W_, ww = l % W_;
  const float* src = xz + ((size_t)b * 2 * DI_ + d) * L_;
  float s = dwb[d];
#pragma unroll
  for (int i = 0; i < 3; ++i) {
    int hh = h + i - 1;
    if (hh < 0 || hh >= H_) continue;
#pragma unroll
    for (int j = 0; j < 3; ++j) {
      int wc = ww + j - 1;
      if (wc < 0 || wc >= W_) continue;
      s += dww[(d * 3 + i) * 3 + j] * src[hh * W_ + wc];
    }
  }
  xc[idx] = siluf_(s);
}

// ======================= x_proj GEMM per direction: xdbl[b,k,c,t] = sum_d xs_k[d,t] xw[k,c,d]
__global__ void k_xproj(const float* __restrict__ xc, const float* __restrict__ xw,
                        float* __restrict__ xdbl) {
  const int lane = threadIdx.x & 31, wave = threadIdx.x >> 5;
  const int half = lane >> 4, l16 = lane & 15;
  const int n0 = (blockIdx.x * 4 + wave) * 64;   // 4 scan-position tiles
  const int m0 = blockIdx.y * 16;                // c tile (48 rows, valid<36)
  const int z  = blockIdx.z;
  const int b = z >> 2, k = z & 3;
  const float* xcb = xc + (size_t)b * DI_ * L_;
  const int c0 = m0 + l16;
  int pix[4];
#pragma unroll
  for (int jt = 0; jt < 4; ++jt) pix[jt] = pixmap(k, n0 + 16 * jt + l16);
  v8f acc0 = {}, acc1 = {}, acc2 = {}, acc3 = {};
  for (int k0 = 0; k0 < DI_; k0 += 4) {
    const int ka = k0 + 2 * half;
    v2f a;
    a.x = (c0 < CDIM_) ? xw[((size_t)k * CDIM_ + c0) * DI_ + ka]     : 0.0f;
    a.y = (c0 < CDIM_) ? xw[((size_t)k * CDIM_ + c0) * DI_ + ka + 1] : 0.0f;
    const float* r0 = xcb + (size_t)ka * L_;
    const float* r1 = xcb + (size_t)(ka + 1) * L_;
    v2f b0, b1, b2, b3;
    b0.x = r0[pix[0]]; b0.y = r1[pix[0]];
    b1.x = r0[pix[1]]; b1.y = r1[pix[1]];
    b2.x = r0[pix[2]]; b2.y = r1[pix[2]];
    b3.x = r0[pix[3]]; b3.y = r1[pix[3]];
    acc0 = wmma_f32_k4(a, b0, acc0);
    acc1 = wmma_f32_k4(a, b1, acc1);
    acc2 = wmma_f32_k4(a, b2, acc2);
    acc3 = wmma_f32_k4(a, b3, acc3);
  }
  v8f* accs[4] = { &acc0, &acc1, &acc2, &acc3 };
#pragma unroll
  for (int jt = 0; jt < 4; ++jt) {
    v8f& a = *accs[jt];
#pragma unroll
    for (int i = 0; i < 8; ++i) {
      int c = m0 + 8 * half + i;
      if (c < CDIM_)
        xdbl[(((size_t)b * KD_ + k) * CDIM_ + c) * L_ + n0 + 16 * jt + l16] = a[i];
    }
  }
}

// ======================= chunked selective scan, pass 1: per-chunk (prod dA, partial h)
__global__ void k_scan1(const float* __restrict__ xc, const float* __restrict__ xdbl,
                        const float* __restrict__ dtw, const float* __restrict__ dtb,
                        const float* __restrict__ alog,
                        float* __restrict__ aprod, float* __restrict__ hpart) {
  __shared__ float sxd[(RR_ + NS_) * CS_];      // dts rows 0..3, B rows 4..19
  const int d = threadIdx.x;
  const int c = blockIdx.x % NC_;
  const int k = (blockIdx.x / NC_) & 3;
  const int b = blockIdx.x / (NC_ * KD_);
  const int t0 = c * CS_;
  stage_tile_async(sxd, xdbl + ((size_t)b * KD_ + k) * CDIM_ * L_ + t0, RR_ + NS_);
  float A[NS_];
#pragma unroll
  for (int n = 0; n < NS_; ++n) A[n] = -__expf(alog[((size_t)(k * DI_ + d)) * NS_ + n]);
  const float w0 = dtw[(k * DI_ + d) * RR_ + 0], w1 = dtw[(k * DI_ + d) * RR_ + 1];
  const float w2 = dtw[(k * DI_ + d) * RR_ + 2], w3 = dtw[(k * DI_ + d) * RR_ + 3];
  const float dbias = dtb[k * DI_ + d];
  const float* xcb = xc + ((size_t)b * DI_ + d) * L_;
  float st[NS_], ap[NS_];
#pragma unroll
  for (int n = 0; n < NS_; ++n) { st[n] = 0.0f; ap[n] = 1.0f; }
  __syncthreads();
  for (int tt = 0; tt < CS_; ++tt) {
    const float u  = xcb[pixmap(k, t0 + tt)];
    const float dl = softplusf_(w0 * sxd[0 * CS_ + tt] + w1 * sxd[1 * CS_ + tt] +
                                w2 * sxd[2 * CS_ + tt] + w3 * sxd[3 * CS_ + tt] + dbias);
    const float du = dl * u;
#pragma unroll
    for (int n = 0; n < NS_; ++n) {
      const float dA = __expf(dl * A[n]);
      st[n] = st[n] * dA + du * sxd[(RR_ + n) * CS_ + tt];
      ap[n] *= dA;
    }
  }
  const size_t base = ((((size_t)b * KD_ + k) * DI_ + d) * NC_ + c) * NS_;
#pragma unroll
  for (int n = 0; n < NS_; ++n) { aprod[base + n] = ap[n]; hpart[base + n] = st[n]; }
}

// pass 2: sequential combine over 72 chunk summaries -> chunk initial states
__global__ void k_scan2(const float* __restrict__ aprod, const float* __restrict__ hpart,
                        float* __restrict__ hinit) {
  const int d = threadIdx.x;
  const int k = blockIdx.x & 3, b = blockIdx.x >> 2;
  const size_t base0 = (((size_t)b * KD_ + k) * DI_ + d) * NC_ * NS_;
  float h[NS_];
#pragma unroll
  for (int n = 0; n < NS_; ++n) h[n] = 0.0f;
  for (int c = 0; c < NC_; ++c) {
    const size_t base = base0 + (size_t)c * NS_;
#pragma unroll
    for (int n = 0; n < NS_; ++n) {
      hinit[base + n] = h[n];
      h[n] = h[n] * aprod[base + n] + hpart[base + n];
    }
  }
}

// pass 3: replay chunk from seeded state, emit ys at original pixel (+ Ds*u)
__global__ void k_scan3(const float* __restrict__ xc, const float* __restrict__ xdbl,
                        const float* __restrict__ dtw, const float* __restrict__ dtb,
                        const float* __restrict__ alog, const float* __restrict__ Dsw,
                        const float* __restrict__ hinit, float* __restrict__ ys) {
  __shared__ float sxd[CDIM_ * CS_];            // dts, B, C rows
  const int d = threadIdx.x;
  const int c = blockIdx.x % NC_;
  const int k = (blockIdx.x / NC_) & 3;
  const int b = blockIdx.x / (NC_ * KD_);
  const int t0 = c * CS_;
  stage_tile_async(sxd, xdbl + ((size_t)b * KD_ + k) * CDIM_ * L_ + t0, CDIM_);
  float A[NS_];
#pragma unroll
  for (int n = 0; n < NS_; ++n) A[n] = -__expf(alog[((size_t)(k * DI_ + d)) * NS_ + n]);
  const float w0 = dtw[(k * DI_ + d) * RR_ + 0], w1 = dtw[(k * DI_ + d) * RR_ + 1];
  const float w2 = dtw[(k * DI_ + d) * RR_ + 2], w3 = dtw[(k * DI_ + d) * RR_ + 3];
  const float dbias = dtb[k * DI_ + d];
  const float Dd = Dsw[k * DI_ + d];
  const float* xcb = xc + ((size_t)b * DI_ + d) * L_;
  float* ysb = ys + (((size_t)b * KD_ + k) * DI_ + d) * L_;
  const size_t base = ((((size_t)b * KD_ + k) * DI_ + d) * NC_ + c) * NS_;
  float st[NS_];
#pragma unroll
  for (int n = 0; n < NS_; ++n) st[n] = hinit[base + n];
  __syncthreads();
  for (int tt = 0; tt < CS_; ++tt) {
    const int pix = pixmap(k, t0 + tt);
    const float u  = xcb[pix];
    const float dl = softplusf_(w0 * sxd[0 * CS_ + tt] + w1 * sxd[1 * CS_ + tt] +
                                w2 * sxd[2 * CS_ + tt] + w3 * sxd[3 * CS_ + tt] + dbias);
    const float du = dl * u;
    float y = 0.0f;
#pragma unroll
    for (int n = 0; n < NS_; ++n) {
      const float dA = __expf(dl * A[n]);
      st[n] = st[n] * dA + du * sxd[(RR_ + n) * CS_ + tt];
      y += st[n] * sxd[(RR_ + NS_ + n) * CS_ + tt];
    }
    ysb[pix] = y + Dd * u;
  }
}

// ======================= combine 4 directions + LN(128) + *silu(z) -> yn[b,pix,d]
__global__ void k_combine_ln(const float* __restrict__ ys, const float* __restrict__ xz,
                             const float* __restrict__ g, const float* __restrict__ bta,
                             float* __restrict__ yn) {
  int row = blockIdx.x * (blockDim.x >> 5) + (threadIdx.x >> 5);
  if (row >= B_ * L_) return;
  int b = row / L_, pix = row % L_;
  int lane = threadIdx.x & 31;
  const float* ysb = ys + (size_t)b * KD_ * DI_ * L_;
  float v[4];
#pragma unroll
  for (int i = 0; i < 4; ++i) {
    int d = lane + 32 * i;
    const float* p = ysb + (size_t)d * L_ + pix;
    v[i] = p[0] + p[(size_t)DI_ * L_] + p[2 * (size_t)DI_ * L_] + p[3 * (size_t)DI_ * L_];
  }
  float s = v[0] + v[1] + v[2] + v[3];
  for (int o = 16; o >= 1; o >>= 1) s += __shfl_xor(s, o, 32);
  float mu = s * (1.0f / 128.0f);
  float q = 0.0f;
#pragma unroll
  for (int i = 0; i < 4; ++i) { float dd = v[i] - mu; q += dd * dd; }
  for (int o = 16; o >= 1; o >>= 1) q += __shfl_xor(q, o, 32);
  float inv = rsqrtf(q * (1.0f / 128.0f) + EPS_);
#pragma unroll
  for (int i = 0; i < 4; ++i) {
    int d = lane + 32 * i;
    float z = xz[((size_t)b * 2 * DI_ + DI_ + d) * L_ + pix];
    yn[((size_t)b * L_ + pix) * DI_ + d] = ((v[i] - mu) * inv * g[d] + bta[d]) * siluf_(z);
  }
}

// ======================= out_proj GEMM + accumulate into S[b,o,pix]
__global__ void k_outproj(const float* __restrict__ yn, const float* __restrict__ w,
                          float* __restrict__ S, int accum) {
  const int lane = threadIdx.x & 31, wave = threadIdx.x >> 5;
  const int half = lane >> 4, l16 = lane & 15;
  const int n0 = (blockIdx.x * 4 + wave) * 64;   // 4 pix tiles
  const int m0 = blockIdx.y * 16;                // o tile
  const int b  = blockIdx.z;
  v8f acc0 = {}, acc1 = {}, acc2 = {}, acc3 = {};
  for (int k0 = 0; k0 < DI_; k0 += 4) {
    const int ka = k0 + 2 * half;
    v2f a;
    a.x = w[(m0 + l16) * DI_ + ka];
    a.y = w[(m0 + l16) * DI_ + ka + 1];
    const float* yb = yn + ((size_t)b * L_ + n0 + l16) * DI_ + ka;
    v2f b0, b1, b2, b3;
    b0.x = yb[0];        b0.y = yb[1];
    b1.x = yb[16 * DI_]; b1.y = yb[16 * DI_ + 1];
    b2.x = yb[32 * DI_]; b2.y = yb[32 * DI_ + 1];
    b3.x = yb[48 * DI_]; b3.y = yb[48 * DI_ + 1];
    acc0 = wmma_f32_k4(a, b0, acc0);
    acc1 = wmma_f32_k4(a, b1, acc1);
    acc2 = wmma_f32_k4(a, b2, acc2);
    acc3 = wmma_f32_k4(a, b3, acc3);
  }
  v8f* accs[4] = { &acc0, &acc1, &acc2, &acc3 };
#pragma unroll
  for (int jt = 0; jt < 4; ++jt) {
    v8f& a = *accs[jt];
#pragma unroll
    for (int i = 0; i < 8; ++i) {
      int m = m0 + 8 * half + i;
      size_t idx = ((size_t)b * HID_ + m) * L_ + n0 + 16 * jt + l16;
      S[idx] = accum ? (S[idx] + a[i]) : a[i];
    }
  }
}

// ======================= out0 = 2*(detail+semantic)*sigmoid(S*S)
__global__ void k_gate(const float* __restrict__ det, const float* __restrict__ sem,
                       const float* __restrict__ S, float* __restrict__ out0) {
  size_t idx = (size_t)blockIdx.x * blockDim.x + threadIdx.x;
  if (idx >= (size_t)B_ * HID_ * L_) return;
  float o = det[idx] + sem[idx];
  float s = S[idx];
  out0[idx] = 2.0f * o * sigmoidf_(s * s);
}

// ======================= conv1 GEMM + sigmoid + *(detail+semantic) -> d_out
__global__ void k_conv1_final(const float* __restrict__ x, const float* __restrict__ w,
                              const float* __restrict__ bias, const float* __restrict__ det,
                              const float* __restrict__ sem, float* __restrict__ out) {
  const int lane = threadIdx.x & 31, wave = threadIdx.x >> 5;
  const int half = lane >> 4, l16 = lane & 15;
  const int n0 = (blockIdx.x * 4 + wave) * 64;   // 4 pix tiles
  const int m0 = blockIdx.y * 16;                // o tile
  const int b  = blockIdx.z;
  const float* xb = x + (size_t)b * HID_ * L_;
  v8f acc0 = {}, acc1 = {}, acc2 = {}, acc3 = {};
  for (int k0 = 0; k0 < HID_; k0 += 4) {
    const int ka = k0 + 2 * half;
    v2f a;
    a.x = w[(m0 + l16) * HID_ + ka];
    a.y = w[(m0 + l16) * HID_ + ka + 1];
    const float* r0 = xb + (size_t)ka * L_ + n0 + l16;
    const float* r1 = xb + (size_t)(ka + 1) * L_ + n0 + l16;
    v2f b0, b1, b2, b3;
    b0.x = r0[0];  b0.y = r1[0];
    b1.x = r0[16]; b1.y = r1[16];
    b2.x = r0[32]; b2.y = r1[32];
    b3.x = r0[48]; b3.y = r1[48];
    acc0 = wmma_f32_k4(a, b0, acc0);
    acc1 = wmma_f32_k4(a, b1, acc1);
    acc2 = wmma_f32_k4(a, b2, acc2);
    acc3 = wmma_f32_k4(a, b3, acc3);
  }
  v8f* accs[4] = { &acc0, &acc1, &acc2, &acc3 };
#pragma unroll
  for (int jt = 0; jt < 4; ++jt) {
    v8f& a = *accs[jt];
#pragma unroll
    for (int i = 0; i < 8; ++i) {
      int m = m0 + 8 * half + i;
      size_t idx = ((size_t)b * HID_ + m) * L_ + n0 + 16 * jt + l16;
      float tv = sigmoidf_(a[i] + bias[m]);
      out[idx] = tv * (det[idx] + sem[idx]);
    }
  }
}

extern "C" void kernel_launch(void* const* d_in, const int* in_sizes, int n_in,
                              void* d_out, int out_size, void* d_ws, size_t ws_size,
                              hipStream_t stream) {
  (void)in_sizes; (void)n_in; (void)out_size; (void)ws_size;
  const float* details    = (const float*)d_in[0];
  const float* semantics  = (const float*)d_in[1];
  const float* conv_w     = (const float*)d_in[2];
  const float* conv_b     = (const float*)d_in[3];
  const float* bn_g       = (const float*)d_in[4];
  const float* bn_b       = (const float*)d_in[5];
  const float* conv1_w    = (const float*)d_in[6];
  const float* conv1_b    = (const float*)d_in[7];
  const float* pe_w       = (const float*)d_in[8];
  const float* pe_b       = (const float*)d_in[9];
  const float* pe_ln_g    = (const float*)d_in[10];
  const float* pe_ln_b    = (const float*)d_in[11];
  const float* in_proj_w  = (const float*)d_in[12];
  const float* dw_w       = (const float*)d_in[13];
  const float* dw_b       = (const float*)d_in[14];
  const float* x_proj_w   = (const float*)d_in[15];
  const float* dt_proj_w  = (const float*)d_in[16];
  const float* dt_proj_b  = (const float*)d_in[17];
  const float* A_log      = (const float*)d_in[18];
  const float* Ds         = (const float*)d_in[19];
  const float* ss_ln_g    = (const float*)d_in[20];
  const float* ss_ln_b    = (const float*)d_in[21];
  const float* out_proj_w = (const float*)d_in[22];
  float* out = (float*)d_out;

  // workspace partition (floats); total ~136 MB
  float* p = (float*)d_ws;
  auto take = [&](size_t n) { float* r = p; p += n; return r; };
  float* detail   = take((size_t)B_ * HID_ * L_);
  float* semantic = take((size_t)B_ * HID_ * L_);
  float* Sb       = take((size_t)B_ * HID_ * L_);
  float* out0     = take((size_t)B_ * HID_ * L_);
  float* yraw     = take((size_t)B_ * L_ * HID_);
  float* xz       = take((size_t)B_ * 2 * DI_ * L_);
  float* xc       = take((size_t)B_ * DI_ * L_);
  float* xdbl     = take((size_t)B_ * KD_ * CDIM_ * L_);
  float* yn       = take((size_t)B_ * L_ * DI_);
  float* ysb      = take((size_t)B_ * KD_ * DI_ * L_);
  float* aprod    = take((size_t)B_ * KD_ * DI_ * NC_ * NS_);
  float* hpart    = take((size_t)B_ * KD_ * DI_ * NC_ * NS_);
  float* hinit    = take((size_t)B_ * KD_ * DI_ * NC_ * NS_);

  const dim3 blk(128);                        // 4 waves; each wave = 16x64 of output
  const dim3 gHID(L_ / 256, HID_ / 16, B_);   // 36 x 6 x 2

  k_conv_bn_silu<<<gHID, blk, 0, stream>>>(details,   conv_w, conv_b, bn_g, bn_b, detail);
  k_conv_bn_silu<<<gHID, blk, 0, stream>>>(semantics, conv_w, conv_b, bn_g, bn_b, semantic);

  const float* srcs[2] = { semantic, detail };   // S = ss(semantic) + ss(detail)
  for (int pass = 0; pass < 2; ++pass) {
    const float* T = srcs[pass];
    k_pe_gemm<<<gHID, blk, 0, stream>>>(T, pe_w, pe_b, yraw);
    k_ln96<<<dim3((B_ * L_ + 7) / 8), dim3(256), 0, stream>>>(yraw, pe_ln_g, pe_ln_b, B_ * L_);
    k_inproj<<<dim3(L_ / 256, 2 * DI_ / 16, B_), blk, 0, stream>>>(yraw, in_proj_w, xz);
    k_dwconv<<<dim3((B_ * DI_ * L_ + 255) / 256), dim3(256), 0, stream>>>(xz, dw_w, dw_b, xc);
    k_xproj<<<dim3(L_ / 256, 3, B_ * KD_), blk, 0, stream>>>(xc, x_proj_w, xdbl);
    k_scan1<<<dim3(B_ * KD_ * NC_), dim3(DI_), 0, stream>>>(xc, xdbl, dt_proj_w, dt_proj_b,
                                                            A_log, aprod, hpart);
    k_scan2<<<dim3(B_ * KD_), dim3(DI_), 0, stream>>>(aprod, hpart, hinit);
    k_scan3<<<dim3(B_ * KD_ * NC_), dim3(DI_), 0, stream>>>(xc, xdbl, dt_proj_w, dt_proj_b,
                                                            A_log, Ds, hinit, ysb);
    k_combine_ln<<<dim3((B_ * L_ + 7) / 8), dim3(256), 0, stream>>>(ysb, xz, ss_ln_g, ss_ln_b, yn);
    k_outproj<<<gHID, blk, 0, stream>>>(yn, out_proj_w, Sb, pass);
  }

  k_gate<<<dim3((B_ * HID_ * L_ + 255) / 256), dim3(256), 0, stream>>>(detail, semantic, Sb, out0);
  k_conv1_final<<<gHID, blk, 0, stream>>>(out0, conv1_w, conv1_b, detail, semantic, out);
}